// QuantumConv2d_82274393522413
// MI455X (gfx1250) — compile-verified
//
#include <hip/hip_runtime.h>

typedef float v2f __attribute__((ext_vector_type(2)));
typedef float v8f __attribute__((ext_vector_type(8)));

#define BATCH_   64
#define SIZE_    256
#define NPR_     128            // patch rows
#define NPC_     128            // patch cols
#define NPCG_    8              // groups of 16 patch-cols
#define NTILES_  (BATCH_ * NPR_ * NPCG_)   // 65536 wave-tiles of 16 patches

__device__ __forceinline__ v8f wmma4(v2f a, v2f b, v8f c) {
  // D = A(16x4 f32) x B(4x16 f32) + C(16x16 f32)
  return __builtin_amdgcn_wmma_f32_16x16x4_f32(false, a, false, b, (short)0, c,
                                               false, false);
}

__global__ __launch_bounds__(256)
void qconv2d_wmma_kernel(const float* __restrict__ x,
                         const float* __restrict__ w,
                         float* __restrict__ out) {
  const int lane = threadIdx.x & 31;
  const int p    = lane & 15;                 // patch slot (N column) / M row
  const int off  = (lane & 16) ? 2 : 0;       // K-local offset per ISA layout

  // ---- Per-wave constant: complex M = CNOT_ring @ kron(RX(w3..w0)) ----
  float cq[4], sq[4];
#pragma unroll
  for (int q = 0; q < 4; ++q) {
    __sincosf(0.5f * w[q], &sq[q], &cq[q]);
  }
  // row s of M = row j=pi^{-1}(s) of u (CNOT ring is a basis permutation)
  const int s0 = p & 1, s1 = (p >> 1) & 1, s2 = (p >> 2) & 1, s3 = (p >> 3) & 1;
  const int j = (s0 ^ s3) | ((s0 ^ s1 ^ s3) << 1) | ((s1 ^ s2) << 2) | ((s2 ^ s3) << 3);

  v2f aR[4], aI[4];   // A-operands (real/imag of M) for the 4 K-chunks
#pragma unroll
  for (int c = 0; c < 4; ++c) {
    float mr[2], mi[2];
#pragma unroll
    for (int jj = 0; jj < 2; ++jj) {
      const int k  = 4 * c + off + jj;
      const int xo = j ^ k;
      const int pc = __popc(xo);
      float prod = 1.0f;
#pragma unroll
      for (int q = 0; q < 4; ++q) prod *= ((xo >> q) & 1) ? sq[q] : cq[q];
      // u[j,k] = (-i)^pc * prod
      if (pc & 1) { mr[jj] = 0.0f; mi[jj] = (pc & 2) ? prod : -prod; }
      else        { mi[jj] = 0.0f; mr[jj] = (pc & 2) ? -prod : prod; }
    }
    aR[c] = (v2f){mr[0], mr[1]};
    aI[c] = (v2f){mi[0], mi[1]};
  }

  // ---- Tile loop: each wave processes 16 patches per tile ----
  const int nWaves = gridDim.x * (blockDim.x >> 5);
  const int gw     = blockIdx.x * (blockDim.x >> 5) + (threadIdx.x >> 5);

  for (int tile = gw; tile < NTILES_; tile += nWaves) {
    const int b   = tile >> 10;        // / (NPR_ * NPCG_)
    const int rem = tile & 1023;
    const int pr  = rem >> 3;
    const int pcg = rem & 7;

    // load the 4 angles of patch (pr, pcg*16 + p)  [both lane halves need them]
    const float* base = x + ((size_t)b * SIZE_ + 2 * pr) * SIZE_ + (pcg * 16 + p) * 2;
    const float2 lo = *(const float2*)base;
    const float2 hi = *(const float2*)(base + SIZE_);
    const float a0 = lo.x, a1 = lo.y, a2 = hi.x, a3 = hi.y;
    const float ss = a0 * a0 + a1 * a1 + a2 * a2 + a3 * a3;

    v8f Cr = {0.f, 0.f, 0.f, 0.f, 0.f, 0.f, 0.f, 0.f};
    v8f Ci = {0.f, 0.f, 0.f, 0.f, 0.f, 0.f, 0.f, 0.f};

#pragma unroll
    for (int c = 0; c < 4; ++c) {
      float cs[2], sn[2];
#pragma unroll
      for (int jj = 0; jj < 2; ++jj) {
        const int k = 4 * c + off + jj;
        const float t = ((k & 1) ? a0 : -a0) + ((k & 2) ? a1 : -a1) +
                        ((k & 4) ? a2 : -a2) + ((k & 8) ? a3 : -a3);
        const float phi = 0.5f * t + 0.25f * (t * t - ss);  // RZ + RZZ phase
        __sincosf(phi, &sn[jj], &cs[jj]);                   // d = cos - i sin
      }
      const v2f bC  = (v2f){cs[0], cs[1]};
      const v2f bS  = (v2f){sn[0], sn[1]};
      const v2f bSn = (v2f){-sn[0], -sn[1]};
      // Cr = Mr@cos + Mi@sin ; Ci = Mi@cos - Mr@sin
      Cr = wmma4(aR[c], bC,  Cr);
      Cr = wmma4(aI[c], bS,  Cr);
      Ci = wmma4(aI[c], bC,  Ci);
      Ci = wmma4(aR[c], bSn, Ci);
    }

    // probs -> per-qubit expectations (STATES is MSB-first bit pick)
    float acc0 = 0.f, acc1 = 0.f, acc2 = 0.f, acc3 = 0.f;
    const int sbase = (lane & 16) ? 8 : 0;
#pragma unroll
    for (int v = 0; v < 8; ++v) {
      const float pv = 0.0625f * (Cr[v] * Cr[v] + Ci[v] * Ci[v]);
      const int s = sbase + v;
      if (s & 8) acc0 += pv;
      if (s & 4) acc1 += pv;
      if (s & 2) acc2 += pv;
      if (s & 1) acc3 += pv;
    }
    acc0 += __shfl_xor(acc0, 16, 32);
    acc1 += __shfl_xor(acc1, 16, 32);
    acc2 += __shfl_xor(acc2, 16, 32);
    acc3 += __shfl_xor(acc3, 16, 32);

    if (lane < 16) {
      float4 o = make_float4(acc0, acc1, acc2, acc3);
      *(float4*)(out + (((size_t)b * NPR_ + pr) * NPC_ + pcg * 16 + p) * 4) = o;
    }
  }
}

extern "C" void kernel_launch(void* const* d_in, const int* in_sizes, int n_in,
                              void* d_out, int out_size, void* d_ws, size_t ws_size,
                              hipStream_t stream) {
  (void)in_sizes; (void)n_in; (void)d_ws; (void)ws_size; (void)out_size;
  const float* x = (const float*)d_in[0];   // [64,256,256] f32
  const float* w = (const float*)d_in[1];   // [2,2] f32
  float* out = (float*)d_out;               // [64,128,128,4] f32
  // 2048 blocks x 8 waves = 16384 waves; 4 tiles (64 patches) per wave
  qconv2d_wmma_kernel<<<dim3(2048), dim3(256), 0, stream>>>(x, w, out);
}